// DotAttention_16673063043141
// MI455X (gfx1250) — compile-verified
//
#include <hip/hip_runtime.h>
#include <math.h>

// Problem constants (from reference)
#define NUM_LAYERS 2
#define Bsz 32
#define Ssz 4096
#define Hsz 1024
#define NEG_INF   (-1.0e9f)

// Tiling
#define ROWS_PER_WAVE 16
#define WAVES_PER_WG  8
#define ROWS_PER_WG   (ROWS_PER_WAVE * WAVES_PER_WG)   // 128
#define CHUNKS        (Ssz / ROWS_PER_WG)              // 32 chunks per batch
#define NPART         (Bsz * CHUNKS)                   // 1024 partial records

// Workspace layout (float offsets)
#define WS_SCORES 0
#define WS_M      (Bsz * Ssz)                  // 131072
#define WS_L      (WS_M + NPART)               // +1024
#define WS_CTX    (WS_L + NPART)               // +1024 ; then NPART*Hsz floats

typedef __attribute__((ext_vector_type(2))) float v2f;
typedef __attribute__((ext_vector_type(8))) float v8f;

// ---------------------------------------------------------------------------
// Pass 1: fused scores + per-chunk softmax partials + partial context.
// One workgroup = (batch b, chunk of 128 S-rows). One wave = 16 S-rows.
// enc is read once from HBM (coalesced float4); the 16-row re-read for the
// context accumulation hits WGP$/L2 (just-touched 64KB working set).
// ---------------------------------------------------------------------------
__global__ __launch_bounds__(256)
void attn_pass1(const float* __restrict__ hidden,
                const float* __restrict__ enc,
                const unsigned char* __restrict__ mask,   // jnp.bool_ -> 1 byte
                float* __restrict__ ws)
{
    __shared__ __align__(16) float ldsP[WAVES_PER_WG * ROWS_PER_WAVE * 32]; // 16 KB
    __shared__ float ldsScore[WAVES_PER_WG * ROWS_PER_WAVE];                // 128 f
    __shared__ float ldsM[WAVES_PER_WG];
    __shared__ float ldsL[WAVES_PER_WG];
    __shared__ __align__(16) float ldsAcc[Hsz];                             // 4 KB

    const int tid   = threadIdx.x;
    const int wave  = tid >> 5;
    const int lane  = tid & 31;
    const int b     = blockIdx.x / CHUNKS;
    const int chunk = blockIdx.x % CHUNKS;
    const int s0    = chunk * ROWS_PER_WG + wave * ROWS_PER_WAVE;

    const float* __restrict__ encB = enc + (size_t)b * Ssz * Hsz;
    const float* __restrict__ hv =
        hidden + ((size_t)(NUM_LAYERS - 1) * Bsz + b) * Hsz;

    // h[b] slice for this lane: 32 floats at h = j*128 + lane*4 + k
    float4 hreg[8];
#pragma unroll
    for (int j = 0; j < 8; ++j)
        hreg[j] = *(const float4*)(hv + j * 128 + lane * 4);

    // ---- Phase 1a: per-lane partial dots for 16 rows (single HBM read) ----
    float* myP = ldsP + wave * (ROWS_PER_WAVE * 32);
#pragma unroll
    for (int r = 0; r < ROWS_PER_WAVE; ++r) {
        const float* row = encB + (size_t)(s0 + r) * Hsz;
        if (r + 2 < ROWS_PER_WAVE)
            __builtin_prefetch(row + 2 * Hsz + lane * 16, 0, 0);
        float d = 0.0f;
#pragma unroll
        for (int j = 0; j < 8; ++j) {
            float4 e = *(const float4*)(row + j * 128 + lane * 4);
            d += e.x * hreg[j].x + e.y * hreg[j].y +
                 e.z * hreg[j].z + e.w * hreg[j].w;
        }
        myP[r * 32 + lane] = d;
    }

    // ---- Phase 1b: cross-lane reduce of P[16][32] via WMMA, B = ones ----
    // A(16x4 f32) layout: lane<16 holds A[m][0..1], lane>=16 holds A[m][2..3].
    // With all-ones B, D[r][n] = sum_lane P[r][lane] for every n.
    const int mrow = lane & 15;
    const int kb   = (lane >> 4) * 2;
    v8f D = {0.f, 0.f, 0.f, 0.f, 0.f, 0.f, 0.f, 0.f};
    v2f ones; ones.x = 1.0f; ones.y = 1.0f;
#pragma unroll
    for (int g = 0; g < 8; ++g) {   // 8 x K=4 covers 32 lanes
        v2f a = *(const v2f*)(myP + mrow * 32 + g * 4 + kb);
        D = __builtin_amdgcn_wmma_f32_16x16x4_f32(
                /*neg_a=*/false, a, /*neg_b=*/false, ones,
                /*c_mod=*/(short)0, D, /*reuse_a=*/false, /*reuse_b=*/false);
    }
    // D VGPR r: lanes 0-15 carry score[r], lanes 16-31 carry score[r+8].
    union { v8f v; float f[8]; } du; du.v = D;
    if (lane == 0) {
#pragma unroll
        for (int r = 0; r < 8; ++r) ldsScore[wave * 16 + r] = du.f[r];
    } else if (lane == 16) {
#pragma unroll
        for (int r = 0; r < 8; ++r) ldsScore[wave * 16 + 8 + r] = du.f[r];
    }

    // ---- Phase 2: masked wave-local softmax stats ----
    const size_t mbase = (size_t)b * Ssz + s0;
    float bm = -INFINITY;
    float ms[ROWS_PER_WAVE];
#pragma unroll
    for (int r = 0; r < ROWS_PER_WAVE; ++r) {
        float sc = ldsScore[wave * 16 + r];
        ms[r] = (mask[mbase + r] != 0) ? sc : NEG_INF;
        bm = fmaxf(bm, ms[r]);
    }
    // persist masked scores for the attn output pass (lane r -> row r)
    if (lane < ROWS_PER_WAVE) {
        float sc = ldsScore[wave * 16 + lane];
        ws[WS_SCORES + mbase + lane] = (mask[mbase + lane] != 0) ? sc : NEG_INF;
    }

    float p[ROWS_PER_WAVE];
    float l = 0.0f;
#pragma unroll
    for (int r = 0; r < ROWS_PER_WAVE; ++r) {
        p[r] = __expf(ms[r] - bm);
        l += p[r];
    }

    // ---- Phase 3: partial context (re-read of hot rows -> cache hits) ----
    float4 acc[8];
#pragma unroll
    for (int j = 0; j < 8; ++j) { acc[j].x = acc[j].y = acc[j].z = acc[j].w = 0.f; }
#pragma unroll
    for (int r = 0; r < ROWS_PER_WAVE; ++r) {
        const float* row = encB + (size_t)(s0 + r) * Hsz;
#pragma unroll
        for (int j = 0; j < 8; ++j) {
            float4 e = *(const float4*)(row + j * 128 + lane * 4);
            acc[j].x += p[r] * e.x; acc[j].y += p[r] * e.y;
            acc[j].z += p[r] * e.z; acc[j].w += p[r] * e.w;
        }
    }

    // ---- Phase 4: combine 8 waves -> one workgroup partial ----
    if (lane == 0) { ldsM[wave] = bm; ldsL[wave] = l; }
#pragma unroll
    for (int k = 0; k < Hsz / 256; ++k) ldsAcc[k * 256 + tid] = 0.0f;
    __syncthreads();

    float M = -INFINITY;
#pragma unroll
    for (int w = 0; w < WAVES_PER_WG; ++w) M = fmaxf(M, ldsM[w]);
    float Lt = 0.0f;
#pragma unroll
    for (int w = 0; w < WAVES_PER_WG; ++w) Lt += __expf(ldsM[w] - M) * ldsL[w];

    const float f = __expf(bm - M);     // this wave's rescale factor
#pragma unroll
    for (int j = 0; j < 8; ++j) {
        const int h = j * 128 + lane * 4;
        atomicAdd(&ldsAcc[h + 0], f * acc[j].x);   // ds_add_f32
        atomicAdd(&ldsAcc[h + 1], f * acc[j].y);
        atomicAdd(&ldsAcc[h + 2], f * acc[j].z);
        atomicAdd(&ldsAcc[h + 3], f * acc[j].w);
    }
    __syncthreads();

    const int pidx = blockIdx.x;        // b * CHUNKS + chunk
    if (tid == 0) { ws[WS_M + pidx] = M; ws[WS_L + pidx] = Lt; }
    *(float4*)(ws + WS_CTX + (size_t)pidx * Hsz + tid * 4) =
        *(const float4*)(ldsAcc + tid * 4);
}

// ---------------------------------------------------------------------------
// Pass 2: combine the 32 chunk-partials per batch; emit context and attn.
// Tiny traffic (~4.7 MB), negligible vs pass 1.
// ---------------------------------------------------------------------------
__global__ __launch_bounds__(256)
void attn_pass2(float* __restrict__ out, const float* __restrict__ ws)
{
    const int b   = blockIdx.x;
    const int tid = threadIdx.x;
    const float* wm = ws + WS_M + b * CHUNKS;
    const float* wl = ws + WS_L + b * CHUNKS;

    float M = -INFINITY;
#pragma unroll
    for (int c = 0; c < CHUNKS; ++c) M = fmaxf(M, wm[c]);
    float L = 0.0f;
#pragma unroll
    for (int c = 0; c < CHUNKS; ++c) L += __expf(wm[c] - M) * wl[c];
    const float invL = 1.0f / L;        // L >= 1 always (max element has p=1)

    // context[b, tid*4 .. tid*4+3]
    float4 acc; acc.x = acc.y = acc.z = acc.w = 0.0f;
#pragma unroll 4
    for (int c = 0; c < CHUNKS; ++c) {
        const float fc = __expf(wm[c] - M);
        float4 v = *(const float4*)(ws + WS_CTX +
                                    (size_t)(b * CHUNKS + c) * Hsz + tid * 4);
        acc.x += fc * v.x; acc.y += fc * v.y;
        acc.z += fc * v.z; acc.w += fc * v.w;
    }
    float4 o;
    o.x = acc.x * invL; o.y = acc.y * invL;
    o.z = acc.z * invL; o.w = acc.w * invL;
    *(float4*)(out + (size_t)b * Hsz + tid * 4) = o;

    // attn[b, :] from stored masked scores (NEG_INF -> exp underflows to 0;
    // all-masked rows reproduce the reference's uniform 1/S)
    float* attn = out + (size_t)Bsz * Hsz;
    const float* sc = ws + WS_SCORES + (size_t)b * Ssz;
#pragma unroll
    for (int it = 0; it < Ssz / 256; ++it) {
        const int s = it * 256 + tid;
        attn[(size_t)b * Ssz + s] = __expf(sc[s] - M) * invL;
    }
}

// ---------------------------------------------------------------------------
extern "C" void kernel_launch(void* const* d_in, const int* in_sizes, int n_in,
                              void* d_out, int out_size, void* d_ws, size_t ws_size,
                              hipStream_t stream)
{
    (void)in_sizes; (void)n_in; (void)out_size; (void)ws_size;
    const float*         hidden = (const float*)d_in[0];          // [L,B,H] f32
    const float*         enc    = (const float*)d_in[1];          // [B,S,H] f32
    const unsigned char* mask   = (const unsigned char*)d_in[2];  // [B,S] bool
    float* out = (float*)d_out;   // context [B,H] ++ attn [B,S]
    float* ws  = (float*)d_ws;    // ~4.75 MB used

    attn_pass1<<<Bsz * CHUNKS, 256, 0, stream>>>(hidden, enc, mask, ws);
    attn_pass2<<<Bsz, 256, 0, stream>>>(out, ws);
}